// QPLayer_3968549782381
// MI455X (gfx1250) — compile-verified
//
#include <hip/hip_runtime.h>

// QP dual projected-gradient, one wave32 per problem.
// M = 2*A*A^T built with V_WMMA_F32_16X16X4_F32 (K=4 == F), iteration is
// register-resident matvec with LDS lambda broadcast.

typedef float v2f __attribute__((ext_vector_type(2)));
typedef float v8f __attribute__((ext_vector_type(8)));

#define QP_ITERS 400
#define MSTRIDE  36   // 32 + 4 pad, keeps rows float4-aligned in LDS

__global__ __launch_bounds__(32) void qp_pgd_kernel(const float* __restrict__ A,
                                                    const float* __restrict__ X,
                                                    const float* __restrict__ Bvec,
                                                    float* __restrict__ Out) {
    __shared__ float sM[32 * MSTRIDE];
    __shared__ float sLam[32];

    const int b    = blockIdx.x;
    const int lane = threadIdx.x;          // 0..31, wave32
    const int r    = lane & 15;
    const bool hi  = lane >= 16;

    // Each lane loads two A rows (row r and row r+16), float4 each.
    const float4* Ab = (const float4*)(A + (size_t)b * 128);
    const float4 rlo = Ab[r];
    const float4 rhi = Ab[r + 16];
    const float4 a   = hi ? rhi : rlo;     // this lane's own constraint row A[lane][:]

    // WMMA f32 16x16x4 operands. For a Gram product the A-operand layout
    // (16x4: lanes 0-15 -> K0,K1; lanes 16-31 -> K2,K3) coincides with the
    // B-operand layout (4x16: K halves split by lane half, N = lane%16), so
    // one v2f per 16-row block serves both sides.
    v2f op0, op1;
    op0[0] = hi ? rlo.z : rlo.x;  op0[1] = hi ? rlo.w : rlo.y;   // rows 0..15
    op1[0] = hi ? rhi.z : rhi.x;  op1[1] = hi ? rhi.w : rhi.y;   // rows 16..31
    const v2f op0s = op0 * 2.0f;  // fold M = 2*A*A^T into the A-side operand
    const v2f op1s = op1 * 2.0f;

    v8f z = {};
    // D = A x B + C ; 4 tiles cover the full 32x32 M.
    v8f d00 = __builtin_amdgcn_wmma_f32_16x16x4_f32(false, op0s, false, op0, (short)0, z, false, false);
    v8f d01 = __builtin_amdgcn_wmma_f32_16x16x4_f32(false, op0s, false, op1, (short)0, z, false, false);
    v8f d10 = __builtin_amdgcn_wmma_f32_16x16x4_f32(false, op1s, false, op0, (short)0, z, false, false);
    v8f d11 = __builtin_amdgcn_wmma_f32_16x16x4_f32(false, op1s, false, op1, (short)0, z, false, false);

    // Scatter D tiles to LDS in row-major M layout.
    // D layout: VGPR rr -> (row = rr + 8*hi, col = lane%16) within the tile.
    {
        const int rowoff = hi ? 8 : 0;
        #pragma unroll
        for (int rr = 0; rr < 8; ++rr) {
            sM[(rr + rowoff) * MSTRIDE + r]           = d00[rr];
            sM[(rr + rowoff) * MSTRIDE + 16 + r]      = d01[rr];
            sM[(16 + rr + rowoff) * MSTRIDE + r]      = d10[rr];
            sM[(16 + rr + rowoff) * MSTRIDE + 16 + r] = d11[rr];
        }
    }
    __syncthreads();   // single-wave workgroup: waitcnt + NOP barrier

    // Pull this lane's M row into VGPRs (lives there for all 400 iterations).
    float Mr[32];
    {
        const float4* rowp = (const float4*)(&sM[lane * MSTRIDE]);
        #pragma unroll
        for (int k = 0; k < 8; ++k) {
            float4 v = rowp[k];
            Mr[4 * k + 0] = v.x; Mr[4 * k + 1] = v.y;
            Mr[4 * k + 2] = v.z; Mr[4 * k + 3] = v.w;
        }
    }

    // alpha = 1 / (||M||_F + 1e-6): per-lane row sum of squares, then butterfly.
    float ss = 0.0f;
    #pragma unroll
    for (int k = 0; k < 32; ++k) ss = fmaf(Mr[k], Mr[k], ss);
    #pragma unroll
    for (int off = 16; off > 0; off >>= 1) ss += __shfl_xor(ss, off, 32);
    const float alpha = 1.0f / (sqrtf(ss) + 1e-6f);

    // q[m] = sum_f A[m][f]*(-x_f) + sum_f A[m][f]*x_f + b[m]  (faithful to Gp+b)
    const float4 xv = ((const float4*)X)[b];
    float q = 0.0f;
    q = fmaf(a.x, -xv.x, q); q = fmaf(a.y, -xv.y, q);
    q = fmaf(a.z, -xv.z, q); q = fmaf(a.w, -xv.w, q);
    q = fmaf(a.x,  xv.x, q); q = fmaf(a.y,  xv.y, q);
    q = fmaf(a.z,  xv.z, q); q = fmaf(a.w,  xv.w, q);
    q += Bvec[(size_t)b * 32 + lane];

    // Projected gradient: lam = relu(lam - alpha*(M lam + q)), 400 iterations.
    float lam = 0.0f;
    const float4* lamv = (const float4*)sLam;
    for (int it = 0; it < QP_ITERS; ++it) {
        sLam[lane] = lam;                  // 1 ds_store_b32
        __syncthreads();
        float p0 = 0.f, p1 = 0.f, p2 = 0.f, p3 = 0.f;
        #pragma unroll
        for (int k = 0; k < 8; ++k) {      // 8 uniform-address ds_load_b128 (broadcast)
            float4 lv = lamv[k];
            p0 = fmaf(Mr[4 * k + 0], lv.x, p0);
            p1 = fmaf(Mr[4 * k + 1], lv.y, p1);
            p2 = fmaf(Mr[4 * k + 2], lv.z, p2);
            p3 = fmaf(Mr[4 * k + 3], lv.w, p3);
        }
        __syncthreads();
        const float grad = ((p0 + p1) + (p2 + p3)) + q;
        lam = fmaxf(0.0f, fmaf(-alpha, grad, lam));
    }

    // s = A^T lam  (butterfly reduce lam*a over lanes), out = 0.5*((x-s)+(-x-s))
    float4 t;
    t.x = a.x * lam; t.y = a.y * lam; t.z = a.z * lam; t.w = a.w * lam;
    #pragma unroll
    for (int off = 16; off > 0; off >>= 1) {
        t.x += __shfl_xor(t.x, off, 32);
        t.y += __shfl_xor(t.y, off, 32);
        t.z += __shfl_xor(t.z, off, 32);
        t.w += __shfl_xor(t.w, off, 32);
    }
    if (lane == 0) {
        float4 o;
        o.x = 0.5f * ((xv.x - t.x) + (-xv.x - t.x));
        o.y = 0.5f * ((xv.y - t.y) + (-xv.y - t.y));
        o.z = 0.5f * ((xv.z - t.z) + (-xv.z - t.z));
        o.w = 0.5f * ((xv.w - t.w) + (-xv.w - t.w));
        ((float4*)Out)[b] = o;
    }
}

extern "C" void kernel_launch(void* const* d_in, const int* in_sizes, int n_in,
                              void* d_out, int out_size, void* d_ws, size_t ws_size,
                              hipStream_t stream) {
    (void)n_in; (void)out_size; (void)d_ws; (void)ws_size;
    const float* A  = (const float*)d_in[0];   // [B,32,4]
    const float* X  = (const float*)d_in[1];   // [B,4]
    const float* Bv = (const float*)d_in[2];   // [B,32]
    float* Out = (float*)d_out;                // [B,4]
    const int nb = in_sizes[0] / 128;          // B
    qp_pgd_kernel<<<nb, 32, 0, stream>>>(A, X, Bv, Out);
}